// AttentionComm_69552700391573
// MI455X (gfx1250) — compile-verified
//
#include <hip/hip_runtime.h>
#include <math.h>

#define NAGENT 64
#define ZDIM   512
#define DDIM   512

typedef __attribute__((ext_vector_type(2))) float v2f;
typedef __attribute__((ext_vector_type(4))) float v4f;
typedef __attribute__((ext_vector_type(8))) float v8f;

// ---------------------------------------------------------------------------
// Phase 1: Q[n]=Wq[n]@z[n], K[n]=Wk[n]@z[n], V[n]=Wv[n]@z[n]
// Bandwidth-bound (192 MB of single-use weights). Wave-per-row GEMV:
//  - each lane preloads its 16 z values (fixed across rows)
//  - 4 rows in flight per wave -> 16 b128 loads outstanding
//  - nontemporal loads: weights are read exactly once
// ---------------------------------------------------------------------------
__global__ __launch_bounds__(256) void qkv_gemv_kernel(
    const float* __restrict__ z,
    const float* __restrict__ Wq,
    const float* __restrict__ Wk,
    const float* __restrict__ Wv,
    float* __restrict__ qkv /* [3][64][512] */) {
  const int b     = blockIdx.x;       // 0..383
  const int mat   = b >> 7;           // 0..2
  const int rem   = b & 127;
  const int agent = rem >> 1;         // 0..63
  const int half  = rem & 1;          // 0..1 (rows 0-255 / 256-511)

  const float* W = (mat == 0 ? Wq : (mat == 1 ? Wk : Wv)) +
                   (size_t)agent * DDIM * ZDIM;
  const float* zp  = z + agent * ZDIM;
  float* outp      = qkv + (size_t)mat * NAGENT * DDIM + agent * DDIM;

  const int lane = threadIdx.x & 31;
  const int wid  = threadIdx.x >> 5;  // 0..7

  v4f zc[4];
#pragma unroll
  for (int it = 0; it < 4; ++it)
    zc[it] = *(const v4f*)(zp + it * 128 + lane * 4);

  const int rbase = half * 256 + wid * 32;
  for (int g = 0; g < 8; ++g) {
    const int r0 = rbase + g * 4;
    float acc[4] = {0.f, 0.f, 0.f, 0.f};
#pragma unroll
    for (int r = 0; r < 4; ++r) {
      const float* row = W + (size_t)(r0 + r) * ZDIM + lane * 4;
#pragma unroll
      for (int it = 0; it < 4; ++it) {
        v4f w = __builtin_nontemporal_load((const v4f*)(row + it * 128));
        acc[r] += w.x * zc[it].x + w.y * zc[it].y + w.z * zc[it].z + w.w * zc[it].w;
      }
    }
#pragma unroll
    for (int r = 0; r < 4; ++r) {
#pragma unroll
      for (int off = 16; off >= 1; off >>= 1)
        acc[r] += __shfl_xor(acc[r], off, 32);
    }
    if (lane == 0) {
#pragma unroll
      for (int r = 0; r < 4; ++r) outp[r0 + r] = acc[r];
    }
  }
}

// ---------------------------------------------------------------------------
// Phase 2: S = (K @ Q^T)/sqrt(D), mask diagonal, softmax over senders (axis 0)
// 16 waves, one 16x16 S-tile per wave via V_WMMA_F32_16X16X4_F32 (exact f32).
// A/B operand layout per ISA 16x4 f32 tables: lane -> row (lane&15),
// K-pair (lane>>4)*2 -> contiguous float2 loads from K / Q rows.
// ---------------------------------------------------------------------------
__global__ __launch_bounds__(512) void scores_softmax_kernel(
    const float* __restrict__ qkv, float* __restrict__ alpha /* [64][64] i-major */) {
  __shared__ float S[NAGENT * NAGENT];
  const float* Q = qkv;
  const float* K = qkv + NAGENT * DDIM;

  const int lane = threadIdx.x & 31;
  const int wid  = threadIdx.x >> 5;       // 0..15
  const int i0   = (wid >> 2) * 16;        // sender tile
  const int j0   = (wid & 3) * 16;         // receiver tile
  const int m    = lane & 15;
  const int koff = (lane >> 4) * 2;

  v8f c = {};
  for (int kk = 0; kk < DDIM; kk += 4) {
    v2f a = *(const v2f*)(K + (i0 + m) * DDIM + kk + koff);
    v2f b = *(const v2f*)(Q + (j0 + m) * DDIM + kk + koff);
    c = __builtin_amdgcn_wmma_f32_16x16x4_f32(false, a, false, b, (short)0, c,
                                              false, false);
  }

  const float inv_scale = 0.04419417382415922f;  // 1/sqrt(512)
#pragma unroll
  for (int r = 0; r < 8; ++r) {
    const int i = i0 + r + (lane >> 4) * 8;
    const int j = j0 + m;
    S[i * NAGENT + j] = c[r] * inv_scale;
  }
  __syncthreads();

  // Column-wise softmax over senders i, diagonal excluded. Thread j owns col j.
  if (threadIdx.x < NAGENT) {
    const int j = threadIdx.x;
    float mx = -3.4e38f;
    for (int i = 0; i < NAGENT; ++i)
      if (i != j) mx = fmaxf(mx, S[i * NAGENT + j]);
    float sum = 0.f;
    for (int i = 0; i < NAGENT; ++i)
      if (i != j) sum += expf(S[i * NAGENT + j] - mx);
    const float inv = 1.f / sum;
    for (int i = 0; i < NAGENT; ++i) {
      const float v = (i == j) ? 0.f : expf(S[i * NAGENT + j] - mx) * inv;
      S[i * NAGENT + j] = v;
    }
  }
  __syncthreads();

  for (int idx = threadIdx.x; idx < NAGENT * NAGENT; idx += blockDim.x)
    alpha[idx] = S[idx];
}

// ---------------------------------------------------------------------------
// Phase 3: messages[j,d] = sum_i alpha[i,j] * V[i,d]  (GEMM: M=j, N=d, K=i)
// One 16x16 output tile per wave, 128 waves total.
// ---------------------------------------------------------------------------
__global__ __launch_bounds__(256) void messages_kernel(
    const float* __restrict__ qkv, const float* __restrict__ alpha,
    float* __restrict__ out /* messages [64][512] */) {
  const float* V = qkv + 2 * NAGENT * DDIM;
  const int lane = threadIdx.x & 31;
  const int wid  = threadIdx.x >> 5;
  const int t    = blockIdx.x * 8 + wid;   // 0..127
  const int j0   = (t >> 5) * 16;          // receiver tile (4)
  const int d0   = (t & 31) * 16;          // feature tile (32)
  const int m    = lane & 15;
  const int koff = (lane >> 4) * 2;

  v8f c = {};
  for (int kk = 0; kk < NAGENT; kk += 4) {
    v2f a, b;
    a.x = alpha[(kk + koff) * NAGENT + j0 + m];      // A[m][k] = alpha[k][j0+m]
    a.y = alpha[(kk + koff + 1) * NAGENT + j0 + m];
    b.x = V[(kk + koff) * DDIM + d0 + m];            // B[k][n] = V[k][d0+n]
    b.y = V[(kk + koff + 1) * DDIM + d0 + m];
    c = __builtin_amdgcn_wmma_f32_16x16x4_f32(false, a, false, b, (short)0, c,
                                              false, false);
  }
#pragma unroll
  for (int r = 0; r < 8; ++r) {
    const int j = j0 + r + (lane >> 4) * 8;
    out[j * DDIM + d0 + m] = c[r];
  }
}

// ---------------------------------------------------------------------------
// Phase 4: hat = V @ Wi^T + bi ; loss = mean_{a,zi} (hat - z)^2
// hat[a][zi] = sum_d V[a][d] * Wi[zi][d]. 128 tiles (4 agent-tiles x 32 zi-
// tiles), one per wave; squared-error reduced in-wave then one atomic/lane0.
// ---------------------------------------------------------------------------
__global__ __launch_bounds__(256) void infer_loss_kernel(
    const float* __restrict__ qkv, const float* __restrict__ Wi,
    const float* __restrict__ bi, const float* __restrict__ z,
    float* __restrict__ loss) {
  const float* V = qkv + 2 * NAGENT * DDIM;
  const int lane = threadIdx.x & 31;
  const int wid  = threadIdx.x >> 5;
  const int t    = blockIdx.x * 8 + wid;   // 0..127
  const int a0   = (t >> 5) * 16;          // agent tile (4)
  const int zi0  = (t & 31) * 16;          // z-dim tile (32)
  const int m    = lane & 15;
  const int koff = (lane >> 4) * 2;

  v8f c = {};
  for (int kk = 0; kk < DDIM; kk += 4) {
    v2f a = *(const v2f*)(V + (a0 + m) * DDIM + kk + koff);
    v2f b = *(const v2f*)(Wi + (zi0 + m) * DDIM + kk + koff);
    c = __builtin_amdgcn_wmma_f32_16x16x4_f32(false, a, false, b, (short)0, c,
                                              false, false);
  }

  const int zi   = zi0 + m;
  const float bv = bi[zi];
  float acc = 0.f;
#pragma unroll
  for (int r = 0; r < 8; ++r) {
    const int ag = a0 + r + (lane >> 4) * 8;
    const float e = c[r] + bv - z[ag * ZDIM + zi];
    acc += e * e;
  }
#pragma unroll
  for (int off = 16; off >= 1; off >>= 1)
    acc += __shfl_xor(acc, off, 32);
  if (lane == 0)
    atomicAdd(loss, acc * (1.0f / (float)(NAGENT * ZDIM)));
}

// ---------------------------------------------------------------------------
extern "C" void kernel_launch(void* const* d_in, const int* in_sizes, int n_in,
                              void* d_out, int out_size, void* d_ws,
                              size_t ws_size, hipStream_t stream) {
  const float* z  = (const float*)d_in[0];
  const float* Wq = (const float*)d_in[1];
  const float* Wk = (const float*)d_in[2];
  const float* Wv = (const float*)d_in[3];
  const float* Wi = (const float*)d_in[4];
  const float* bi = (const float*)d_in[5];
  float* out = (float*)d_out;                 // [0..32767]=messages, [32768]=loss

  float* qkv   = (float*)d_ws;                // 3 * 64 * 512 floats (Q|K|V)
  float* alpha = qkv + 3 * NAGENT * DDIM;     // 64 * 64 floats

  // zero the scalar loss accumulator (capture-safe)
  (void)hipMemsetAsync(out + NAGENT * DDIM, 0, sizeof(float), stream);

  qkv_gemv_kernel<<<384, 256, 0, stream>>>(z, Wq, Wk, Wv, qkv);
  scores_softmax_kernel<<<1, 512, 0, stream>>>(qkv, alpha);
  messages_kernel<<<16, 256, 0, stream>>>(qkv, alpha, out);
  infer_loss_kernel<<<16, 256, 0, stream>>>(qkv, Wi, bi, z,
                                            out + NAGENT * DDIM);
}